// InterColumnSynapses_56624848830930
// MI455X (gfx1250) — compile-verified
//
#include <hip/hip_runtime.h>

// InterColumnSynapses for MI455X (gfx1250, wave32, WMMA).
//
// out[t][b][0:64]  = sum_{s<t} l23[s][b] @ W_ff[edge(s,t)]^T
// out[t][b][64:96] = sum_{s>t} l5 [s][b] @ W_fb[edge(s,t)]^T
//
// fp32 path via V_WMMA_F32_16X16X4_F32 (M=16,N=16,K=4), weights staged in LDS.

typedef float v2f __attribute__((ext_vector_type(2)));
typedef float v4f __attribute__((ext_vector_type(4)));
typedef float v8f __attribute__((ext_vector_type(8)));

#define N_COLS  8
#define BATCH   65536
#define NE      32      // n_l23e == n_l5e
#define EMBED   64
#define NOUT    96      // 64 + 32

// Static connectivity: dense off-diagonal. Edge enumeration matches the
// reference's lexicographic (s, t) order.
__device__ __forceinline__ int ff_edge_idx(int s, int t) {
    // edges (s,t) with s < t, ordered by s then t
    return s * 7 - (s * (s - 1)) / 2 + (t - s - 1);
}
__device__ __forceinline__ int fb_edge_idx(int s, int t) {
    // edges (s,t) with s > t, ordered by s then t
    return (s * (s - 1)) / 2 + t;
}

__global__ __launch_bounds__(256)
void icsyn_wmma_kernel(const float* __restrict__ l23,
                       const float* __restrict__ l5,
                       const float* __restrict__ Wff,
                       const float* __restrict__ Wfb,
                       float* __restrict__ out)
{
    // Union LDS buffer: FF weights for t sources (t * 2048 floats) followed by
    // FB weights for (7-t) sources ((7-t) * 1024 floats). Worst case t=7:
    // 14336 floats = 56 KB.
    __shared__ float smem[14336];

    const int t   = blockIdx.y;          // target column, uniform per block
    const int tid = threadIdx.x;

    float* s_ff = smem;                  // slot s (s<t)      : stride 2048 floats
    float* s_fb = smem + t * (EMBED * NE); // slot j=s-(t+1)  : stride 1024 floats

    // ---- Cooperative weight staging into LDS (float4 granularity) ----
    {
        const int nff4 = t * (EMBED * NE / 4);       // t * 512
        for (int i = tid; i < nff4; i += 256) {
            int s   = i >> 9;                        // i / 512
            int off = i & 511;
            const v4f* src = (const v4f*)(Wff + (size_t)ff_edge_idx(s, t) * (EMBED * NE));
            ((v4f*)s_ff)[i] = src[off];
        }
        const int nfb4 = (7 - t) * (NE * NE / 4);    // (7-t) * 256
        for (int i = tid; i < nfb4; i += 256) {
            int j   = i >> 8;                        // i / 256 -> s = t+1+j
            int off = i & 255;
            const v4f* src = (const v4f*)(Wfb + (size_t)fb_edge_idx(t + 1 + j, t) * (NE * NE));
            ((v4f*)s_fb)[i] = src[off];
        }
    }
    __syncthreads();

    const int lane = tid & 31;
    const int wave = tid >> 5;
    const int h    = lane >> 4;      // lane half: selects K pair {2h, 2h+1}
    const int ln   = lane & 15;      // row (A) / column (B,C,D) within tile
    const int rowbase = (blockIdx.x * 8 + wave) * 32;   // 32 batch rows per wave

    v8f acc_th[2][4] = {};           // thalamic: 2 M-tiles x 4 N-tiles (N=64)
    v8f acc_fb[2][2] = {};           // feedback: 2 M-tiles x 2 N-tiles (N=32)

    // ---- Feedforward: sources s < t, K = 32 per source ----
    for (int s = 0; s < t; ++s) {
        const float* x0 = l23 + ((size_t)s * BATCH + rowbase + ln) * NE;       // M-tile 0
        const float* x1 = x0 + (size_t)16 * NE;                                // M-tile 1
        const float* w  = s_ff + s * (EMBED * NE);                             // [64][32] row-major
#pragma unroll
        for (int kk = 0; kk < NE; kk += 4) {
            const int ko = kk + 2 * h;
            // B fragments: B[k][n] = W[n][k]; lane -> n = nt*16 + ln, K pair from lane half
            v2f b0 = *(const v2f*)(w + (0 * 16 + ln) * NE + ko);
            v2f b1 = *(const v2f*)(w + (1 * 16 + ln) * NE + ko);
            v2f b2 = *(const v2f*)(w + (2 * 16 + ln) * NE + ko);
            v2f b3 = *(const v2f*)(w + (3 * 16 + ln) * NE + ko);
            // A fragments: lane -> row = ln, K pair from lane half
            v2f a0 = *(const v2f*)(x0 + ko);
            v2f a1 = *(const v2f*)(x1 + ko);
            acc_th[0][0] = __builtin_amdgcn_wmma_f32_16x16x4_f32(false, a0, false, b0, (short)0, acc_th[0][0], false, false);
            acc_th[0][1] = __builtin_amdgcn_wmma_f32_16x16x4_f32(false, a0, false, b1, (short)0, acc_th[0][1], false, false);
            acc_th[0][2] = __builtin_amdgcn_wmma_f32_16x16x4_f32(false, a0, false, b2, (short)0, acc_th[0][2], false, false);
            acc_th[0][3] = __builtin_amdgcn_wmma_f32_16x16x4_f32(false, a0, false, b3, (short)0, acc_th[0][3], false, false);
            acc_th[1][0] = __builtin_amdgcn_wmma_f32_16x16x4_f32(false, a1, false, b0, (short)0, acc_th[1][0], false, false);
            acc_th[1][1] = __builtin_amdgcn_wmma_f32_16x16x4_f32(false, a1, false, b1, (short)0, acc_th[1][1], false, false);
            acc_th[1][2] = __builtin_amdgcn_wmma_f32_16x16x4_f32(false, a1, false, b2, (short)0, acc_th[1][2], false, false);
            acc_th[1][3] = __builtin_amdgcn_wmma_f32_16x16x4_f32(false, a1, false, b3, (short)0, acc_th[1][3], false, false);
        }
    }

    // ---- Feedback: sources s > t, K = 32 per source ----
    for (int s = t + 1; s < N_COLS; ++s) {
        const float* y0 = l5 + ((size_t)s * BATCH + rowbase + ln) * NE;
        const float* y1 = y0 + (size_t)16 * NE;
        const float* w  = s_fb + (s - t - 1) * (NE * NE);                      // [32][32] row-major
#pragma unroll
        for (int kk = 0; kk < NE; kk += 4) {
            const int ko = kk + 2 * h;
            v2f b0 = *(const v2f*)(w + (0 * 16 + ln) * NE + ko);
            v2f b1 = *(const v2f*)(w + (1 * 16 + ln) * NE + ko);
            v2f a0 = *(const v2f*)(y0 + ko);
            v2f a1 = *(const v2f*)(y1 + ko);
            acc_fb[0][0] = __builtin_amdgcn_wmma_f32_16x16x4_f32(false, a0, false, b0, (short)0, acc_fb[0][0], false, false);
            acc_fb[0][1] = __builtin_amdgcn_wmma_f32_16x16x4_f32(false, a0, false, b1, (short)0, acc_fb[0][1], false, false);
            acc_fb[1][0] = __builtin_amdgcn_wmma_f32_16x16x4_f32(false, a1, false, b0, (short)0, acc_fb[1][0], false, false);
            acc_fb[1][1] = __builtin_amdgcn_wmma_f32_16x16x4_f32(false, a1, false, b1, (short)0, acc_fb[1][1], false, false);
        }
    }

    // ---- Store: C/D layout -> element (M = r + 8*h, N = ln) per VGPR r ----
    float* obase = out + ((size_t)t * BATCH + rowbase) * NOUT;
#pragma unroll
    for (int mi = 0; mi < 2; ++mi) {
#pragma unroll
        for (int nt = 0; nt < 4; ++nt) {
#pragma unroll
            for (int r = 0; r < 8; ++r) {
                obase[(size_t)(mi * 16 + r + 8 * h) * NOUT + nt * 16 + ln] = acc_th[mi][nt][r];
            }
        }
#pragma unroll
        for (int nt = 0; nt < 2; ++nt) {
#pragma unroll
            for (int r = 0; r < 8; ++r) {
                obase[(size_t)(mi * 16 + r + 8 * h) * NOUT + 64 + nt * 16 + ln] = acc_fb[mi][nt][r];
            }
        }
    }
}

extern "C" void kernel_launch(void* const* d_in, const int* in_sizes, int n_in,
                              void* d_out, int out_size, void* d_ws, size_t ws_size,
                              hipStream_t stream) {
    (void)in_sizes; (void)n_in; (void)out_size; (void)d_ws; (void)ws_size;
    const float* l23 = (const float*)d_in[0];
    const float* l5  = (const float*)d_in[1];
    const float* Wff = (const float*)d_in[2];
    const float* Wfb = (const float*)d_in[3];
    // d_in[4..7] (ff_src/ff_tgt/fb_src/fb_tgt) encode the static dense
    // off-diagonal mask; edge indices are computed in closed form on device.
    float* out = (float*)d_out;

    dim3 grid(BATCH / 256, N_COLS);   // 256 batch tiles (8 waves x 32 rows) x 8 targets
    dim3 block(256);
    icsyn_wmma_kernel<<<grid, block, 0, stream>>>(l23, l5, Wff, Wfb, out);
}